// GradeientMCFA_41077067219547
// MI455X (gfx1250) — compile-verified
//
#include <hip/hip_runtime.h>

typedef __attribute__((ext_vector_type(2))) float v2f;
typedef __attribute__((ext_vector_type(8))) float v8f;

#define MDIM 512
#define NPIX (MDIM * MDIM)
#define LBANDS 31
#define LDB 15
#define PIX_PER_BLOCK 128
#define THREADS 256

__device__ __forceinline__ unsigned f2ord(float f) {
  unsigned u = __float_as_uint(f);
  return (u & 0x80000000u) ? ~u : (u | 0x80000000u);
}
__device__ __forceinline__ float ord2f(unsigned o) {
  unsigned u = (o & 0x80000000u) ? (o & 0x7fffffffu) : ~o;
  return __uint_as_float(u);
}

__global__ void mcfa_init_max(unsigned* gmax) {
  if (threadIdx.x == 0) *gmax = 0u;   // ordered-uint "minus infinity"
}

// Pass 1: spectral downsample via f32 WMMA, forward model, residual, adjoint.
// Writes un-normalized Xn to out and the global max (ordered uint) to gmax.
__global__ __launch_bounds__(THREADS) void mcfa_grad_kernel(
    const float* __restrict__ X, const float* __restrict__ y,
    const float* __restrict__ H, float* __restrict__ out,
    unsigned* __restrict__ gmax) {
  __shared__ float Wlds[16][32];          // 15x31 resize matrix, zero padded
  __shared__ float XdL[PIX_PER_BLOCK][17];  // downsampled spectra (pad=17)

  const int tid = threadIdx.x;

  // Build the antialiased-linear (triangle kernel, width 31/15, per-row
  // normalized) resize matrix exactly as jax.image.resize does.
  for (int e = tid; e < 16 * 32; e += THREADS) {
    const int band = e >> 5;
    const int k = e & 31;
    float w = 0.0f;
    if (band < LDB && k < LBANDS) {
      const float ksc = 31.0f / 15.0f;               // kernel scale (downsample)
      const float c = ((float)band + 0.5f) * ksc - 0.5f;
      float s = 0.0f;
      for (int i = 0; i < LBANDS; ++i)
        s += fmaxf(0.0f, 1.0f - fabsf((float)i - c) / ksc);
      w = fmaxf(0.0f, 1.0f - fabsf((float)k - c) / ksc) / s;
    }
    Wlds[band][k] = w;
  }
  __syncthreads();

  const int wave = tid >> 5;      // wave32 on gfx1250
  const int lane = tid & 31;
  const int laneLo = lane & 15;
  const bool hi = lane >= 16;

  // Each wave computes a 16-pixel x 16-band tile: D = A(16x4) * B(4x16), 8 steps.
  const int pixWave = blockIdx.x * PIX_PER_BLOCK + wave * 16;
  const float* xrow = X + (size_t)(pixWave + laneLo) * LBANDS;

  v8f acc = {};
#pragma unroll
  for (int step = 0; step < 8; ++step) {
    const int k0 = step * 4 + (hi ? 2 : 0);          // ISA 32-bit 16x4 A layout
    v2f a, b;
    a.x = xrow[k0];                                   // k0 <= 30 always
    a.y = (k0 + 1 < LBANDS) ? xrow[k0 + 1] : 0.0f;    // pad K=31 with zero
    b.x = Wlds[laneLo][k0];                           // B[k][n] = W[n][k]
    b.y = Wlds[laneLo][k0 + 1];
    acc = __builtin_amdgcn_wmma_f32_16x16x4_f32(
        false, a, false, b, (short)0, acc, false, false);
  }

  // D layout: VGPR v, lanes 0-15 -> row v, lanes 16-31 -> row v+8; col = lane%16.
#pragma unroll
  for (int v = 0; v < 8; ++v) {
    XdL[wave * 16 + v + (hi ? 8 : 0)][laneLo] = acc[v];
  }
  __syncthreads();

  // One thread per pixel: sensing sum, residual, adjoint, local max.
  float lmax = -3.4e38f;
  if (tid < PIX_PER_BLOCK) {
    const size_t p = (size_t)blockIdx.x * PIX_PER_BLOCK + tid;
    const float* hrow = H + p * LDB;
    float h[LDB];
    float s = 0.0f;
#pragma unroll
    for (int l = 0; l < LDB; ++l) {
      h[l] = hrow[l];
      s += h[l] * XdL[tid][l];
    }
    const float r = s - y[p];
    float* orow = out + p * LDB;
#pragma unroll
    for (int l = 0; l < LDB; ++l) {
      const float xn = h[l] * r;
      orow[l] = xn;
      lmax = fmaxf(lmax, xn);
    }
  }

  // Wave-level max reduce (ordered uint), one atomic per active wave.
  unsigned om = f2ord(lmax);
#pragma unroll
  for (int off = 16; off > 0; off >>= 1) {
    unsigned o = (unsigned)__shfl_xor((int)om, off, 32);
    om = om > o ? om : o;
  }
  if (lane == 0 && wave < (PIX_PER_BLOCK / 32)) atomicMax(gmax, om);
}

// Pass 2: divide by global max (float4 grid-stride; count divisible by 4).
__global__ __launch_bounds__(THREADS) void mcfa_normalize(
    float* __restrict__ out, const unsigned* __restrict__ gmax, int n4) {
  const float inv = 1.0f / ord2f(*gmax);
  float4* o4 = (float4*)out;
  for (int i = blockIdx.x * blockDim.x + threadIdx.x; i < n4;
       i += gridDim.x * blockDim.x) {
    float4 v = o4[i];
    v.x *= inv; v.y *= inv; v.z *= inv; v.w *= inv;
    o4[i] = v;
  }
}

extern "C" void kernel_launch(void* const* d_in, const int* in_sizes, int n_in,
                              void* d_out, int out_size, void* d_ws, size_t ws_size,
                              hipStream_t stream) {
  (void)in_sizes; (void)n_in; (void)ws_size;
  const float* X = (const float*)d_in[0];   // (512,512,31)
  const float* y = (const float*)d_in[1];   // (1,512,512,1,1)
  const float* H = (const float*)d_in[2];   // (1,512,512,15,1)
  float* out = (float*)d_out;               // (1,512,512,15)
  unsigned* gmax = (unsigned*)d_ws;

  mcfa_init_max<<<1, 32, 0, stream>>>(gmax);

  const int blocks = NPIX / PIX_PER_BLOCK;  // 2048
  mcfa_grad_kernel<<<blocks, THREADS, 0, stream>>>(X, y, H, out, gmax);

  const int n4 = out_size / 4;              // 3932160 / 4
  int nblk = (n4 + THREADS - 1) / THREADS;
  if (nblk > 4096) nblk = 4096;
  mcfa_normalize<<<nblk, THREADS, 0, stream>>>(out, gmax, n4);
}